// BscanPreproc_31258771980568
// MI455X (gfx1250) — compile-verified
//
#include <hip/hip_runtime.h>
#include <math.h>

// ---------------- constants matching the reference ----------------
#define BN   16
#define HH   512
#define WW   512
#define NPIX (BN*HH*WW)          // 4194304
#define GH   8
#define GW   8
#define TSZ  64                  // tile height/width (512/8)
#define NB   256                 // histogram bins
#define CLIPV 16.0f              // max(1*4096//256, 1)
#define PIXT 4096.0f             // pixels per tile

typedef float v2f __attribute__((ext_vector_type(2)));
typedef float v8f __attribute__((ext_vector_type(8)));

// normalized gauss1d(5, sigma=1): exp(-x^2/2)/sum  (folds to immediates)
constexpr float G1C[5] = {
    0.054488684f, 0.244201347f, 0.402619928f, 0.244201347f, 0.054488684f};

#define DEVI __device__ __forceinline__

// branchless 5-tap band coefficient: g[d] for d in [0,4], else 0
DEVI float bandg(int d) {
    float r = 0.0f;
    r = (d == 0 || d == 4) ? 0.054488684f : r;
    r = (d == 1 || d == 3) ? 0.244201347f : r;
    r = (d == 2)           ? 0.402619928f : r;
    return r;
}

// order-preserving uint key for float min/max via integer atomics
DEVI unsigned fkey(float f) {
    unsigned u = __float_as_uint(f);
    return (u & 0x80000000u) ? ~u : (u | 0x80000000u);
}
DEVI float fdec(unsigned k) {
    unsigned u = (k & 0x80000000u) ? (k ^ 0x80000000u) : ~k;
    return __uint_as_float(u);
}
DEVI int refl(int i, int n) {            // jnp.pad 'reflect'
    if (i < 0) i = -i;
    if (i >= n) i = 2 * n - 2 - i;
    return i;
}

// ---------------- kernel 0: init stats ----------------
__global__ void k_init(unsigned* stats) {
    if (threadIdx.x < 6) stats[threadIdx.x] = (threadIdx.x & 1) ? 0u : 0xFFFFFFFFu;
}

// ---------------- kernel 1: global min/max (b128 loads + prefetch) ----------------
__global__ void k_minmax(const float4* __restrict__ in, unsigned* stats, int n4) {
    __shared__ unsigned smn[256], smx[256];
    unsigned mn = 0xFFFFFFFFu, mx = 0u;
    const int stride = gridDim.x * blockDim.x;
    for (int i = blockIdx.x * blockDim.x + threadIdx.x; i < n4; i += stride) {
        __builtin_prefetch(&in[i + stride], 0, 1);   // global_prefetch_b8
        float4 v = in[i];
        unsigned k0 = fkey(v.x), k1 = fkey(v.y), k2 = fkey(v.z), k3 = fkey(v.w);
        mn = min(min(mn, k0), min(k1, min(k2, k3)));
        mx = max(max(mx, k0), max(k1, max(k2, k3)));
    }
    smn[threadIdx.x] = mn; smx[threadIdx.x] = mx;
    __syncthreads();
    for (int s = 128; s > 0; s >>= 1) {
        if (threadIdx.x < s) {
            smn[threadIdx.x] = min(smn[threadIdx.x], smn[threadIdx.x + s]);
            smx[threadIdx.x] = max(smx[threadIdx.x], smx[threadIdx.x + s]);
        }
        __syncthreads();
    }
    if (threadIdx.x == 0) { atomicMin(&stats[0], smn[0]); atomicMax(&stats[1], smx[0]); }
}

// ---------------- kernel 2: normalize + log-adjust + CLAHE hist/LUT ----------------
// one block (256 thr) per 64x64 tile; 1024 tiles total
__global__ void k_hist(const float* __restrict__ x, float* __restrict__ adj,
                       float* __restrict__ lut, const unsigned* __restrict__ stats) {
    __shared__ unsigned hist[NB];
    __shared__ float s[NB];
    const int t = blockIdx.x;                 // tile id, 0..1023
    const int b = t >> 6;
    const int cell = t & 63;
    const int gy = cell >> 3, gx = cell & 7;
    const float mn = fdec(stats[0]), mx = fdec(stats[1]);
    const float inv = 1.0f / (mx - mn);
    hist[threadIdx.x] = 0u;
    __syncthreads();
    const float* img  = x   + (size_t)b * HH * WW;
    float*       aimg = adj + (size_t)b * HH * WW;
    #pragma unroll
    for (int p = 0; p < 16; ++p) {
        int l  = p * 256 + threadIdx.x;       // row-major within tile -> coalesced
        int ly = l >> 6, lx = l & 63;
        int gidx = (gy * TSZ + ly) * WW + gx * TSZ + lx;
        float v  = (img[gidx] - mn) * inv;
        float xa = fminf(fmaxf(2.5f * __log2f(1.0f + v), 0.0f), 1.0f);
        aimg[gidx] = xa;
        int bin = min(max((int)(xa * 256.0f), 0), 255);
        atomicAdd(&hist[bin], 1u);
    }
    __syncthreads();
    const int i = threadIdx.x;
    s[i] = fminf((float)hist[i], CLIPV);      // clipped histogram
    __syncthreads();
    // Hillis-Steele inclusive scan over 256 bins
    for (int off = 1; off < NB; off <<= 1) {
        float tv = (i >= off) ? s[i - off] : 0.0f;
        __syncthreads();
        s[i] += tv;
        __syncthreads();
    }
    float total    = s[NB - 1];
    float residual = (PIXT - total) / (float)NB;
    float cum      = s[i] + residual * (float)(i + 1);
    lut[(size_t)t * NB + i] =
        floorf(fminf(fmaxf(cum * (255.0f / PIXT), 0.0f), 255.0f));
}

// ---------------- kernel 3: CLAHE bilinear LUT blend (pointwise, in-place) ----------------
__global__ void k_interp(float* __restrict__ adj, const float* __restrict__ lut,
                         unsigned* stats) {
    __shared__ unsigned smn[256], smx[256];
    const int g = blockIdx.x * 256 + threadIdx.x;     // exactly NPIX threads
    const int b = g >> 18;
    const int rem = g & 262143;
    const int y = rem >> 9, xc = rem & 511;
    float xa = adj[g];
    int bin = min(max((int)(xa * 256.0f), 0), 255);
    float py = (y  + 0.5f) * (1.0f / 64.0f) - 0.5f;
    float px = (xc + 0.5f) * (1.0f / 64.0f) - 0.5f;
    int iy0 = (int)floorf(py); float fy = py - (float)iy0;
    int ix0 = (int)floorf(px); float fx = px - (float)ix0;
    int iy0c = min(max(iy0, 0), GH - 1), iy1c = min(max(iy0 + 1, 0), GH - 1);
    int ix0c = min(max(ix0, 0), GW - 1), ix1c = min(max(ix0 + 1, 0), GW - 1);
    const float* lb = lut + (size_t)b * GH * GW * NB;
    float v00 = lb[(iy0c * GW + ix0c) * NB + bin];
    float v01 = lb[(iy0c * GW + ix1c) * NB + bin];
    float v10 = lb[(iy1c * GW + ix0c) * NB + bin];
    float v11 = lb[(iy1c * GW + ix1c) * NB + bin];
    float o = ((1.0f - fy) * ((1.0f - fx) * v00 + fx * v01) +
               fy          * ((1.0f - fx) * v10 + fx * v11)) * (1.0f / 255.0f);
    adj[g] = o;
    unsigned k = fkey(o);
    smn[threadIdx.x] = k; smx[threadIdx.x] = k;
    __syncthreads();
    for (int sft = 128; sft > 0; sft >>= 1) {
        if (threadIdx.x < sft) {
            smn[threadIdx.x] = min(smn[threadIdx.x], smn[threadIdx.x + sft]);
            smx[threadIdx.x] = max(smx[threadIdx.x], smx[threadIdx.x + sft]);
        }
        __syncthreads();
    }
    if (threadIdx.x == 0) { atomicMin(&stats[2], smn[0]); atomicMax(&stats[3], smx[0]); }
}

// ---------------- kernel 4: normalize-on-load + 5x5 bilateral ----------------
__global__ void k_bilateral(const float* __restrict__ in, float* __restrict__ out,
                            const unsigned* __restrict__ stats) {
    __shared__ float s[20 * 20];
    const float mn = fdec(stats[2]), mx = fdec(stats[3]);
    const float inv = 1.0f / (mx - mn);
    const int b  = blockIdx.z;
    const int y0 = blockIdx.y * 16, x0 = blockIdx.x * 16;
    const float* img = in + (size_t)b * HH * WW;
    const int tid = threadIdx.y * 16 + threadIdx.x;
    for (int i = tid; i < 400; i += 256) {
        int ly = i / 20, lx = i % 20;
        int gy = refl(y0 - 2 + ly, HH), gx = refl(x0 - 2 + lx, WW);
        s[i] = (img[gy * WW + gx] - mn) * inv;
    }
    __syncthreads();
    const int ly = threadIdx.y, lx = threadIdx.x;
    const float c = s[(ly + 2) * 20 + lx + 2];
    float sw = 0.0f, sv = 0.0f;
    #pragma unroll
    for (int dy = 0; dy < 5; ++dy) {
        #pragma unroll
        for (int dx = 0; dx < 5; ++dx) {
            float nb = s[(ly + dy) * 20 + lx + dx];
            float d  = nb - c;
            // sigma_color = 0.1 -> -0.5/0.01 = -50
            float w = __expf(-50.0f * d * d) * (G1C[dy] * G1C[dx]);
            sw += w; sv += nb * w;
        }
    }
    out[(size_t)b * HH * WW + (y0 + ly) * WW + x0 + lx] = sv / sw;
}

// ---------------- kernel 5: horizontal 5-tap gaussian via WMMA ----------------
// one wave32 per 16x16 tile: D(16x16) = A(16x20 data) x W(20x16 band), 5x K=4 chains.
// A needs lane m to hold row m (vertical gather) -> stage tile in LDS with a
// coalesced row-major load, then read A operands as conflict-free ds_load_b64.
__global__ void k_gaussH(const float* __restrict__ in, float* __restrict__ out) {
    __shared__ float st[8][16 * 20];          // per-wave 16x20 tile, row stride 20
    const int lane = threadIdx.x & 31;
    const int wv   = threadIdx.x >> 5;
    const int tile = blockIdx.x * 8 + wv;     // 16384 tiles total
    const int b  = tile >> 10;
    const int r  = tile & 1023;
    const int ty = r >> 5, tx = r & 31;
    const int y0 = ty * 16, x0 = tx * 16;
    const float* img = in + (size_t)b * HH * WW;
    float* sm = st[wv];
    // cooperative coalesced stage: consecutive lanes -> consecutive columns
    #pragma unroll
    for (int k = 0; k < 10; ++k) {
        int i  = lane + k * 32;               // 0..319
        int ly = i / 20, lx = i - ly * 20;
        sm[i] = img[(y0 + ly) * WW + refl(x0 - 2 + lx, WW)];
    }
    __syncthreads();
    const int half = (lane < 16) ? 0 : 2;
    const int m = lane & 15;                  // A row / D column index half
    v8f acc = {};
    #pragma unroll
    for (int c = 0; c < 5; ++c) {
        int j0 = 4 * c + half;                // K index of a.x / b.x
        // A = data from LDS: adjacent columns -> single 8B-aligned ds_load_b64
        v2f a = *(const v2f*)&sm[m * 20 + j0];
        v2f bm;
        // B = band coefficients: W[j][n] = g[j-n] for 0<=j-n<5  (branchless selects)
        bm.x = bandg(j0 - m);
        bm.y = bandg(j0 + 1 - m);
        acc = __builtin_amdgcn_wmma_f32_16x16x4_f32(
            false, a, false, bm, (short)0, acc, false, false);
    }
    float* o = out + (size_t)b * HH * WW;
    #pragma unroll
    for (int rr = 0; rr < 8; ++rr) {
        int M = (lane < 16) ? rr : rr + 8;    // D layout: VGPR rr -> rows rr / rr+8
        o[(y0 + M) * WW + x0 + m] = acc[rr];
    }
}

// ---------------- kernel 6: vertical 5-tap gaussian via WMMA + min/max ----------------
// D(16x16) = Wv(16x20 band) x B(20x16 data); B loads are coalesced (64B/half-wave)
__global__ void k_gaussV(const float* __restrict__ in, float* __restrict__ out,
                         unsigned* stats) {
    __shared__ unsigned smn[256], smx[256];
    const int lane = threadIdx.x & 31;
    const int wv   = threadIdx.x >> 5;
    const int tile = blockIdx.x * 8 + wv;
    const int b  = tile >> 10;
    const int r  = tile & 1023;
    const int ty = r >> 5, tx = r & 31;
    const int y0 = ty * 16, x0 = tx * 16;
    const float* img = in + (size_t)b * HH * WW;
    const int half = (lane < 16) ? 0 : 2;
    const int m = lane & 15;                  // A row (coef) / B column (data)
    v8f acc = {};
    #pragma unroll
    for (int c = 0; c < 5; ++c) {
        int j0 = 4 * c + half;
        v2f a, bm;
        // A = band coefficients: Wv[m][j] = g[j-m] for 0<=j-m<5  (branchless selects)
        a.x = bandg(j0 - m);
        a.y = bandg(j0 + 1 - m);
        // B = data: B[j][n] = img[y0-2+j][x0+n] (reflect rows)
        bm.x = img[refl(y0 - 2 + j0,     HH) * WW + x0 + m];
        bm.y = img[refl(y0 - 2 + j0 + 1, HH) * WW + x0 + m];
        acc = __builtin_amdgcn_wmma_f32_16x16x4_f32(
            false, a, false, bm, (short)0, acc, false, false);
    }
    float* o = out + (size_t)b * HH * WW;
    unsigned mnk = 0xFFFFFFFFu, mxk = 0u;
    #pragma unroll
    for (int rr = 0; rr < 8; ++rr) {
        int M = (lane < 16) ? rr : rr + 8;
        float v = acc[rr];
        o[(y0 + M) * WW + x0 + m] = v;
        unsigned k = fkey(v);
        mnk = min(mnk, k); mxk = max(mxk, k);
    }
    smn[threadIdx.x] = mnk; smx[threadIdx.x] = mxk;
    __syncthreads();
    for (int sft = 128; sft > 0; sft >>= 1) {
        if (threadIdx.x < sft) {
            smn[threadIdx.x] = min(smn[threadIdx.x], smn[threadIdx.x + sft]);
            smx[threadIdx.x] = max(smx[threadIdx.x], smx[threadIdx.x + sft]);
        }
        __syncthreads();
    }
    if (threadIdx.x == 0) { atomicMin(&stats[4], smn[0]); atomicMax(&stats[5], smx[0]); }
}

// ---------------- kernel 7: final normalize (b128 load/store) ----------------
__global__ void k_final(const float4* __restrict__ in, float4* __restrict__ out,
                        const unsigned* __restrict__ stats) {
    const int g = blockIdx.x * 256 + threadIdx.x;     // float4 index
    const float mn = fdec(stats[4]);
    const float inv = 1.0f / (fdec(stats[5]) - mn);
    float4 v = in[g];
    float4 o;
    o.x = (v.x - mn) * inv;
    o.y = (v.y - mn) * inv;
    o.z = (v.z - mn) * inv;
    o.w = (v.w - mn) * inv;
    out[g] = o;
}

// ---------------- host-side orchestration ----------------
extern "C" void kernel_launch(void* const* d_in, const int* in_sizes, int n_in,
                              void* d_out, int out_size, void* d_ws, size_t ws_size,
                              hipStream_t stream) {
    (void)in_sizes; (void)n_in; (void)out_size; (void)ws_size;
    const float* x = (const float*)d_in[0];
    float* out = (float*)d_out;

    // workspace layout: bufA | bufB | stats(64B) | lut (1024*256 f32)
    float*    bufA  = (float*)d_ws;
    float*    bufB  = bufA + NPIX;
    unsigned* stats = (unsigned*)(bufB + NPIX);
    float*    lut   = (float*)((char*)stats + 64);

    k_init     <<<1, 64, 0, stream>>>(stats);
    k_minmax   <<<1024, 256, 0, stream>>>((const float4*)x, stats, NPIX / 4);
    k_hist     <<<BN * GH * GW, 256, 0, stream>>>(x, bufA, lut, stats);
    k_interp   <<<NPIX / 256, 256, 0, stream>>>(bufA, lut, stats);
    k_bilateral<<<dim3(WW / 16, HH / 16, BN), dim3(16, 16), 0, stream>>>(bufA, bufB, stats);
    k_gaussH   <<<(BN * 32 * 32) / 8, 256, 0, stream>>>(bufB, bufA);
    k_gaussV   <<<(BN * 32 * 32) / 8, 256, 0, stream>>>(bufA, bufB, stats);
    k_final    <<<NPIX / 1024, 256, 0, stream>>>((const float4*)bufB, (float4*)out, stats);
}